// SGC_79577154060358
// MI455X (gfx1250) — compile-verified
//
#include <hip/hip_runtime.h>

// SGConv (K=3) on MI455X / gfx1250.
//   1) CSR build over dst: count (int atomics) -> single-block scan -> fill
//   2) dinv = rsqrt(indeg + 1)
//   3) 3x gather hop (no value atomics): hn[n] = dinv[n]^2*h[n] + sum_j dinv[n]*dinv[s_j]*h[s_j]
//      ping-pong x -> hA -> hB -> hA (51MB each; both resident in 192MB L2)
//   4) out = hA @ W^T + b via V_WMMA_F32_16X16X4_F32, software-pipelined fragment loads

#define N_NODES 100000
#define N_EDGES 800000
#define DIM     128
#define ROW_BLKS (N_NODES / 16)   // 6250, exact
#define SCAN_B  1024

typedef __attribute__((ext_vector_type(2))) float v2f;
typedef __attribute__((ext_vector_type(8))) float v8f;

// ---------------- CSR build ----------------

__global__ void sgc_zero_cnt(int* __restrict__ cnt) {
    int i = blockIdx.x * blockDim.x + threadIdx.x;
    if (i < N_NODES) cnt[i] = 0;
}

__global__ void sgc_cnt(const int* __restrict__ dst, int* __restrict__ cnt) {
    int e = blockIdx.x * blockDim.x + threadIdx.x;
    if (e < N_EDGES) atomicAdd(&cnt[dst[e]], 1);
}

__global__ void sgc_dinv(const int* __restrict__ cnt, float* __restrict__ dinv) {
    int i = blockIdx.x * blockDim.x + threadIdx.x;
    if (i < N_NODES) dinv[i] = rsqrtf((float)cnt[i] + 1.0f);   // +1 self loop
}

// single-workgroup exclusive scan: rowptr[i] = sum_{j<i} cnt[j]; rowptr[N] = E
__global__ void sgc_scan(const int* __restrict__ cnt, int* __restrict__ rowptr) {
    __shared__ int buf[SCAN_B];
    __shared__ int carry;
    int tid = threadIdx.x;
    if (tid == 0) carry = 0;
    __syncthreads();
    for (int base = 0; base < N_NODES; base += SCAN_B) {
        int i = base + tid;
        int v = (i < N_NODES) ? cnt[i] : 0;
        buf[tid] = v;
        __syncthreads();
        for (int off = 1; off < SCAN_B; off <<= 1) {   // inclusive Hillis-Steele
            int t = (tid >= off) ? buf[tid - off] : 0;
            __syncthreads();
            buf[tid] += t;
            __syncthreads();
        }
        int incl = buf[tid];
        if (i < N_NODES) rowptr[i] = carry + (incl - v);
        __syncthreads();
        if (tid == SCAN_B - 1) carry += incl;
        __syncthreads();
    }
    if (tid == 0) rowptr[N_NODES] = carry;
}

// cursor starts as a copy of rowptr; E index atomics total
__global__ void sgc_csr_fill(const int* __restrict__ src, const int* __restrict__ dst,
                             int* __restrict__ cursor, int* __restrict__ csr_src) {
    int e = blockIdx.x * blockDim.x + threadIdx.x;
    if (e < N_EDGES) {
        int pos = atomicAdd(&cursor[dst[e]], 1);
        csr_src[pos] = src[e];
    }
}

// ---------------- gather hop (one wave32 per node, float4 per lane) ----------------

__global__ void sgc_hop_gather(const float* __restrict__ h, const float* __restrict__ dinv,
                               const int* __restrict__ rowptr, const int* __restrict__ csr_src,
                               float* __restrict__ hn) {
    int n = blockIdx.x * (blockDim.x >> 5) + (threadIdx.x >> 5);
    if (n >= N_NODES) return;                 // wave-uniform
    int lane = threadIdx.x & 31;

    float dn = dinv[n];
    float c0 = dn * dn;                       // self-loop coefficient
    float4 acc = *(const float4*)(h + (size_t)n * DIM + lane * 4);
    acc.x *= c0; acc.y *= c0; acc.z *= c0; acc.w *= c0;

    int beg = rowptr[n];
    int end = rowptr[n + 1];
    for (int j = beg; j < end; ++j) {         // wave-uniform trip count
        int s = csr_src[j];                   // broadcast load
        float c = dn * dinv[s];
        float4 v = *(const float4*)(h + (size_t)s * DIM + lane * 4);
        acc.x = fmaf(c, v.x, acc.x);
        acc.y = fmaf(c, v.y, acc.y);
        acc.z = fmaf(c, v.z, acc.z);
        acc.w = fmaf(c, v.w, acc.w);
    }
    *(float4*)(hn + (size_t)n * DIM + lane * 4) = acc;
}

// ---------------- final linear: out = H @ W^T + b (fp32 WMMA, pipelined) ----------------
// A frag (16x4 f32): lane<16 -> M=lane, K={k0,k0+1}; lane>=16 -> K={k0+2,k0+3}
// B frag (4x16 f32): lane<16 -> N=lane, K={k0,k0+1}; lane>=16 -> K={k0+2,k0+3}; B[k][n]=W[n][k]
// C/D 16x16 f32: VGPR v -> M = v + 8*(lane>=16), N = lane&15
__global__ void sgc_gemm_wmma(const float* __restrict__ H, const float* __restrict__ W,
                              const float* __restrict__ bias, float* __restrict__ out) {
    int wave = blockIdx.x * (blockDim.x >> 5) + (threadIdx.x >> 5);
    if (wave >= ROW_BLKS) return;             // wave-uniform: EXEC all-1s for WMMA
    int lane = threadIdx.x & 31;
    int half = lane >> 4;
    int l16  = lane & 15;
    int m0   = wave * 16;

    v8f acc[8] = {};

    const float* hrow = H + (size_t)(m0 + l16) * DIM + 2 * half;
    const float* wbase = W + (size_t)l16 * DIM + 2 * half;

    // software pipeline: fragments for k-step k+1 load while k's 8 WMMAs issue
    v2f a = *(const v2f*)(hrow);
    v2f bf[8];
#pragma unroll
    for (int t = 0; t < 8; ++t)
        bf[t] = *(const v2f*)(wbase + (size_t)t * 16 * DIM);

#pragma unroll
    for (int k = 0; k < 32; ++k) {
        v2f a_n;
        v2f bf_n[8];
        if (k < 31) {
            int k0n = (k + 1) * 4;
            a_n = *(const v2f*)(hrow + k0n);
#pragma unroll
            for (int t = 0; t < 8; ++t)
                bf_n[t] = *(const v2f*)(wbase + (size_t)t * 16 * DIM + k0n);
        }
#pragma unroll
        for (int t = 0; t < 8; ++t)
            acc[t] = __builtin_amdgcn_wmma_f32_16x16x4_f32(
                false, a, false, bf[t], (short)0, acc[t], false, false);
        if (k < 31) {
            a = a_n;
#pragma unroll
            for (int t = 0; t < 8; ++t) bf[t] = bf_n[t];
        }
    }

#pragma unroll
    for (int t = 0; t < 8; ++t) {
        float bv = bias[16 * t + l16];
#pragma unroll
        for (int v = 0; v < 8; ++v) {
            int row = m0 + v + 8 * half;
            out[(size_t)row * DIM + 16 * t + l16] = acc[t][v] + bv;
        }
    }
}

// ---------------- launch ----------------

extern "C" void kernel_launch(void* const* d_in, const int* in_sizes, int n_in,
                              void* d_out, int out_size, void* d_ws, size_t ws_size,
                              hipStream_t stream) {
    (void)in_sizes; (void)n_in; (void)out_size; (void)ws_size;

    const float* x   = (const float*)d_in[0];   // [N,128]
    const int*   ei  = (const int*)d_in[1];     // [2,E]: [0]=src, [1]=dst
    const float* W   = (const float*)d_in[2];   // [128,128]
    const float* b   = (const float*)d_in[3];   // [128]
    float*       out = (float*)d_out;           // [N,128]

    const int* src = ei;
    const int* dst = ei + N_EDGES;

    // workspace layout (4-byte units):
    //   dinv[102400] | rowptr[102400] | cursor[102400] | csr_src[800000] | hA | hB  (~107 MB)
    float* dinv    = (float*)d_ws;
    int*   rowptr  = (int*)d_ws + 102400;
    int*   cursor  = (int*)d_ws + 204800;      // also serves as cnt[] during build
    int*   csr_src = (int*)d_ws + 307200;
    float* hA      = (float*)d_ws + 1107200;   // 256B-aligned
    float* hB      = hA + (size_t)N_NODES * DIM;

    const int B = 256;
    const int gN = (N_NODES + B - 1) / B;
    const int gE = (N_EDGES + B - 1) / B;

    int* cnt = cursor;                          // reuse: cnt lives in cursor buffer
    sgc_zero_cnt<<<gN, B, 0, stream>>>(cnt);
    sgc_cnt     <<<gE, B, 0, stream>>>(dst, cnt);
    sgc_dinv    <<<gN, B, 0, stream>>>(cnt, dinv);
    sgc_scan    <<<1, SCAN_B, 0, stream>>>(cnt, rowptr);
    // cursor <- rowptr (overwrites cnt, which is no longer needed)
    hipMemcpyAsync(cursor, rowptr, (size_t)N_NODES * sizeof(int),
                   hipMemcpyDeviceToDevice, stream);
    sgc_csr_fill<<<gE, B, 0, stream>>>(src, dst, cursor, csr_src);

    // hops: one wave32 per node, 8 waves per 256-thread block
    const int hopGrid = (N_NODES + 7) / 8;
    sgc_hop_gather<<<hopGrid, B, 0, stream>>>(x,  dinv, rowptr, csr_src, hA);
    sgc_hop_gather<<<hopGrid, B, 0, stream>>>(hA, dinv, rowptr, csr_src, hB);
    sgc_hop_gather<<<hopGrid, B, 0, stream>>>(hB, dinv, rowptr, csr_src, hA);

    // fp32 WMMA linear: 8 waves/block, 6250 row tiles
    const int gemmGrid = (ROW_BLKS + 7) / 8;
    sgc_gemm_wmma<<<gemmGrid, B, 0, stream>>>(hA, W, b, out);
}